// GGSAGE_18554258719174
// MI455X (gfx1250) — compile-verified
//
#include <hip/hip_runtime.h>
#include <math.h>

#define N_NODES 100000
#define N_EDGES 1600000
#define D 64

typedef __attribute__((ext_vector_type(2))) float v2f;
typedef __attribute__((ext_vector_type(8))) float v8f;

// ---------------------------------------------------------------------------
// Zero out aggregation + degree buffers (grid-stride).
// ---------------------------------------------------------------------------
__global__ void GGSAGE_zero_kernel(float* __restrict__ p, int n) {
  for (int i = blockIdx.x * blockDim.x + threadIdx.x; i < n;
       i += gridDim.x * blockDim.x)
    p[i] = 0.0f;
}

// ---------------------------------------------------------------------------
// Edge-parallel mean aggregation: one wave32 per edge.
// Lanes cover channels [0..31] and [32..63]; coalesced row loads from X,
// f32 atomic adds into AGG[dst], lane 0 counts degree. L2-atomic bound.
// ---------------------------------------------------------------------------
__global__ __launch_bounds__(256) void GGSAGE_scatter_kernel(
    const float* __restrict__ X, const long long* __restrict__ SRC,
    const long long* __restrict__ DST, float* __restrict__ AGG,
    float* __restrict__ DEG, int E) {
  int wid = (int)((blockIdx.x * blockDim.x + threadIdx.x) >> 5);
  int lane = threadIdx.x & 31;
  if (wid >= E) return;
  long long s = SRC[wid];
  long long d = DST[wid];
  const float* xs = X + (size_t)s * D;
  float* ad = AGG + (size_t)d * D;
  atomicAdd(&ad[lane], xs[lane]);
  atomicAdd(&ad[lane + 32], xs[lane + 32]);
  if (lane == 0) atomicAdd(&DEG[d], 1.0f);
}

// ---------------------------------------------------------------------------
// Fused SAGE layer: OUT[16x64 tile] = act( l2norm_rows( (AGG/deg)@WL^T + b
//                                                      + X@WR^T ) )
// One wave32 per 16-node tile, fp32 WMMA 16x16x4.
//   A layout (32-bit 16x4): lane = M (0..15 per half), half-wave selects
//     K pair {0,1} vs {2,3}; v2f per lane.
//   B layout (4x16, mirrors A): lane = N, half-wave selects K pair.
//   C/D layout (16x16 f32): VGPR v of lanes 0..15 = row v; lanes 16..31 =
//     row v+8; N = lane&15 (+16*ntile).
// act: 0 = relu, 1 = elu
// ---------------------------------------------------------------------------
__global__ __launch_bounds__(256) void GGSAGE_wmma_kernel(
    const float* __restrict__ X, const float* __restrict__ AGG,
    const float* __restrict__ DEG, const float* __restrict__ WL,
    const float* __restrict__ BL, const float* __restrict__ WR,
    float* __restrict__ OUT, int n_nodes, int act) {
  // Weights staged K-major so B-fragment LDS reads are (mostly) conflict-free.
  __shared__ float lwlT[D * D];  // lwlT[k*64 + n] = WL[n*64 + k]
  __shared__ float lwrT[D * D];
  __shared__ float lb[D];

  for (int i = threadIdx.x; i < D * D; i += blockDim.x) {
    int n = i >> 6;
    int k = i & 63;
    lwlT[k * D + n] = WL[i];
    lwrT[k * D + n] = WR[i];
  }
  if (threadIdx.x < D) lb[threadIdx.x] = BL[threadIdx.x];
  __syncthreads();

  int wave = blockIdx.x * (blockDim.x >> 5) + (threadIdx.x >> 5);
  int lane = threadIdx.x & 31;
  int tile0 = wave * 16;
  if (tile0 >= n_nodes) return;  // wave-uniform: EXEC stays all-ones

  int m = lane & 15;            // M for A-frags, N for B/C-frags
  int khalf = (lane >> 4) << 1; // 0 for lanes 0-15, 2 for lanes 16-31
  int row = tile0 + m;

  float rdeg = 1.0f / fmaxf(DEG[row], 1.0f);

  // Preload both A operands (agg/deg and x) for all 16 K-steps.
  v2f aL[16], aR[16];
  {
    const float* ap = AGG + (size_t)row * D;
    const float* xp = X + (size_t)row * D;
#pragma unroll
    for (int kk = 0; kk < 16; ++kk) {
      int k = kk * 4 + khalf;
      aL[kk].x = ap[k] * rdeg;
      aL[kk].y = ap[k + 1] * rdeg;
      aR[kk].x = xp[k];
      aR[kk].y = xp[k + 1];
    }
  }

  // Accumulators: 4 N-tiles of 16x16, seeded with the bias row.
  v8f acc[4];
#pragma unroll
  for (int nt = 0; nt < 4; ++nt) {
    float b = lb[nt * 16 + m];
    acc[nt] = (v8f){b, b, b, b, b, b, b, b};
  }

#pragma unroll
  for (int nt = 0; nt < 4; ++nt) {
    int n = nt * 16 + m;
#pragma unroll
    for (int kk = 0; kk < 16; ++kk) {
      int k = kk * 4 + khalf;
      v2f bL, bR;
      bL.x = lwlT[k * D + n];
      bL.y = lwlT[(k + 1) * D + n];
      bR.x = lwrT[k * D + n];
      bR.y = lwrT[(k + 1) * D + n];
      acc[nt] = __builtin_amdgcn_wmma_f32_16x16x4_f32(
          false, aL[kk], false, bL, (short)0, acc[nt], false, false);
      acc[nt] = __builtin_amdgcn_wmma_f32_16x16x4_f32(
          false, aR[kk], false, bR, (short)0, acc[nt], false, false);
    }
  }

  // Row-wise L2 norm: component v lives in a fixed row per half-wave, so
  // reduce squares across the 16 lanes of each half (xor masks < 16).
  float ss[8];
#pragma unroll
  for (int v = 0; v < 8; ++v) {
    float s2 = 0.0f;
#pragma unroll
    for (int nt = 0; nt < 4; ++nt) {
      float val = acc[nt][v];
      s2 += val * val;
    }
#pragma unroll
    for (int mask = 1; mask < 16; mask <<= 1) s2 += __shfl_xor(s2, mask, 32);
    ss[v] = s2;
  }

  int row_base = tile0 + ((lane >> 4) << 3);  // +v gives the output row
#pragma unroll
  for (int v = 0; v < 8; ++v) {
    float rn = 1.0f / fmaxf(sqrtf(ss[v]), 1e-12f);
    size_t orow = (size_t)(row_base + v) * D;
#pragma unroll
    for (int nt = 0; nt < 4; ++nt) {
      float val = acc[nt][v] * rn;
      if (act == 0)
        val = fmaxf(val, 0.0f);                       // relu
      else
        val = (val > 0.0f) ? val : expm1f(val);       // elu (alpha = 1)
      OUT[orow + nt * 16 + m] = val;
    }
  }
}

// ---------------------------------------------------------------------------
// Launch: zero -> scatter -> wmma(layer1,relu) -> zero -> scatter(h)
//         -> wmma(layer2,elu)
// ---------------------------------------------------------------------------
extern "C" void kernel_launch(void* const* d_in, const int* in_sizes, int n_in,
                              void* d_out, int out_size, void* d_ws,
                              size_t ws_size, hipStream_t stream) {
  const float* x = (const float*)d_in[0];
  const long long* ei = (const long long*)d_in[1];
  const long long* src = ei;
  const long long* dst = ei + N_EDGES;
  const float* w1l = (const float*)d_in[2];
  const float* b1l = (const float*)d_in[3];
  const float* w1r = (const float*)d_in[4];
  const float* w2l = (const float*)d_in[5];
  const float* b2l = (const float*)d_in[6];
  const float* w2r = (const float*)d_in[7];
  float* out = (float*)d_out;

  float* agg = (float*)d_ws;                     // 100000*64 floats
  float* deg = agg + (size_t)N_NODES * D;        // 100000 floats
  float* h = deg + N_NODES;                      // 100000*64 floats

  const int zn = N_NODES * D + N_NODES;          // agg+deg contiguous
  const int sblocks = N_EDGES / 8;               // wave per edge, 8 waves/block
  const int tiles = (N_NODES + 15) / 16;         // 6250
  const int gblocks = (tiles + 7) / 8;           // 8 tiles (waves) per block

  // Layer 1
  GGSAGE_zero_kernel<<<2048, 256, 0, stream>>>(agg, zn);
  GGSAGE_scatter_kernel<<<sblocks, 256, 0, stream>>>(x, src, dst, agg, deg,
                                                     N_EDGES);
  GGSAGE_wmma_kernel<<<gblocks, 256, 0, stream>>>(x, agg, deg, w1l, b1l, w1r,
                                                  h, N_NODES, 0);
  // Layer 2
  GGSAGE_zero_kernel<<<2048, 256, 0, stream>>>(agg, zn);
  GGSAGE_scatter_kernel<<<sblocks, 256, 0, stream>>>(h, src, dst, agg, deg,
                                                     N_EDGES);
  GGSAGE_wmma_kernel<<<gblocks, 256, 0, stream>>>(h, agg, deg, w2l, b2l, w2r,
                                                  out, N_NODES, 1);
}